// NsNet2_70592082477279
// MI455X (gfx1250) — compile-verified
//
#include <hip/hip_runtime.h>
#include <hip/hip_bf16.h>
#include <math.h>

// ---------------------------------------------------------------------------
// NsNet2 single-step inference for MI455X (gfx1250, wave32).
// GEMVs run on V_WMMA_F32_16X16X4_F32 (full fp32 matrix pipe):
//   - one 16-row output tile per wave, K consumed 4 per WMMA
//   - A (16x4): lane L holds W[m0+L][k0+2*hi .. +1]  (hi = lane>>4)
//   - B (4x16): x[k0..k0+3] broadcast to all 16 columns (same lane pattern)
//   - D column 0 -> lanes 0 (M=m0..m0+7) and 16 (M=m0+8..m0+15)
// Hot loop has no selects: row index clamped once, K tail as masked epilogue.
// ---------------------------------------------------------------------------

typedef __attribute__((ext_vector_type(2))) float v2f;
typedef __attribute__((ext_vector_type(8))) float v8f;

#define H400 400

// ACT: 0 = none, 1 = relu, 2 = sigmoid
template <int ACT>
__device__ __forceinline__ void gemv16_tile(const float* __restrict__ W,
                                            const float* __restrict__ x,
                                            const float* __restrict__ bias,
                                            float* __restrict__ y,
                                            int rows, int K, int m0) {
    const int lane = threadIdx.x & 31;
    const int hi   = lane >> 4;                 // 0: K {0,1}, 1: K {2,3}
    int row = m0 + (lane & 15);
    row = (row < rows) ? row : (rows - 1);      // clamp: extra rows never stored

    const float* __restrict__ wr = W + (size_t)row * (size_t)K + 2 * hi;
    const float* __restrict__ xp = x + 2 * hi;
    const int K4 = K & ~3;

    v8f acc = {};
#pragma unroll 4
    for (int k = 0; k < K4; k += 4) {
        v2f a, b;
        a.x = wr[k];
        a.y = wr[k + 1];
        b.x = xp[k];
        b.y = xp[k + 1];
        acc = __builtin_amdgcn_wmma_f32_16x16x4_f32(
            false, a, false, b, (short)0, acc, false, false);
    }

    if (K4 != K) {                              // masked epilogue (<=3 elems)
        const int e0 = K4 + 2 * hi;
        v2f a = {0.0f, 0.0f}, b = {0.0f, 0.0f};
        if (e0 < K)     { a.x = wr[K4];     b.x = xp[K4];     }
        if (e0 + 1 < K) { a.y = wr[K4 + 1]; b.y = xp[K4 + 1]; }
        acc = __builtin_amdgcn_wmma_f32_16x16x4_f32(
            false, a, false, b, (short)0, acc, false, false);
    }

    // Column N=0: lane 0 holds M=m0..m0+7, lane 16 holds M=m0+8..m0+15.
    if ((lane & 15) == 0) {
        const int base = m0 + hi * 8;
#pragma unroll
        for (int v = 0; v < 8; ++v) {
            const int m = base + v;
            if (m < rows) {
                float r = acc[v] + (bias ? bias[m] : 0.0f);
                if (ACT == 1)      r = fmaxf(r, 0.0f);
                else if (ACT == 2) r = 1.0f / (1.0f + expf(-r));
                y[m] = r;
            }
        }
    }
}

template <int ACT>
__global__ void gemv_wmma(const float* __restrict__ W, const float* __restrict__ x,
                          const float* __restrict__ bias, float* __restrict__ y,
                          int rows, int K) {
    const int wave = (blockIdx.x * blockDim.x + threadIdx.x) >> 5;
    const int m0   = wave * 16;
    if (m0 >= rows) return;                     // wave-uniform
    gemv16_tile<ACT>(W, x, bias, y, rows, K, m0);
}

// Two independent GEMVs (Wi@x and Wh@h of one GRU cell) in a single launch.
__global__ void gemv_dual_wmma(const float* __restrict__ W0, const float* __restrict__ x0,
                               float* __restrict__ y0,
                               const float* __restrict__ W1, const float* __restrict__ x1,
                               float* __restrict__ y1,
                               int rows, int K) {
    const int wave  = (blockIdx.x * blockDim.x + threadIdx.x) >> 5;
    const int tiles = (rows + 15) / 16;
    if (wave >= 2 * tiles) return;              // wave-uniform
    if (wave < tiles)                           // wave-uniform branch
        gemv16_tile<0>(W0, x0, nullptr, y0, rows, K, wave * 16);
    else
        gemv16_tile<0>(W1, x1, nullptr, y1, rows, K, (wave - tiles) * 16);
}

// GRU gate combine (ONNX zrn order): h' = (1-z)*n + z*h
__global__ void gru_gate_kernel(const float* __restrict__ gi,  // [3H] = Wi @ x
                                const float* __restrict__ gh,  // [3H] = Wh @ h
                                const float* __restrict__ b,   // [6H]
                                const float* __restrict__ h,   // [H]
                                float* __restrict__ hn) {      // [H]
    const int j = blockIdx.x * blockDim.x + threadIdx.x;
    if (j >= H400) return;
    const float zi = gi[j]            + b[j];
    const float ri = gi[H400 + j]     + b[H400 + j];
    const float ni = gi[2 * H400 + j] + b[2 * H400 + j];
    const float zh = gh[j]            + b[3 * H400 + j];
    const float rh = gh[H400 + j]     + b[4 * H400 + j];
    const float nh = gh[2 * H400 + j] + b[5 * H400 + j];
    const float r = 1.0f / (1.0f + expf(-(ri + rh)));
    const float z = 1.0f / (1.0f + expf(-(zi + zh)));
    const float n = tanhf(ni + r * nh);
    hn[j] = (1.0f - z) * n + z * h[j];
}

static inline int blocks_for_waves(int waves) {
    return (waves * 32 + 255) / 256;
}
static inline int gemv_blocks(int rows) {
    return blocks_for_waves((rows + 15) / 16);
}

extern "C" void kernel_launch(void* const* d_in, const int* in_sizes, int n_in,
                              void* d_out, int out_size, void* d_ws, size_t ws_size,
                              hipStream_t stream) {
    (void)in_sizes; (void)n_in; (void)out_size; (void)ws_size;

    const float* x       = (const float*)d_in[0];   // [257]
    const float* h1      = (const float*)d_in[1];   // [400]
    const float* h2      = (const float*)d_in[2];   // [400]
    const float* fc1_w   = (const float*)d_in[3];   // [400,257]
    const float* fc1_b   = (const float*)d_in[4];   // [400]
    const float* gru1_wi = (const float*)d_in[5];   // [1200,400]
    const float* gru1_wh = (const float*)d_in[6];   // [1200,400]
    const float* gru1_b  = (const float*)d_in[7];   // [2400]
    const float* gru2_wi = (const float*)d_in[8];   // [1200,400]
    const float* gru2_wh = (const float*)d_in[9];   // [1200,400]
    const float* gru2_b  = (const float*)d_in[10];  // [2400]
    const float* fc2_w   = (const float*)d_in[11];  // [600,400]
    const float* fc2_b   = (const float*)d_in[12];  // [600]
    const float* fc3_w   = (const float*)d_in[13];  // [600,600]
    const float* fc3_b   = (const float*)d_in[14];  // [600]
    const float* fc4_w   = (const float*)d_in[15];  // [257,600]
    const float* fc4_b   = (const float*)d_in[16];  // [257]
    float* out = (float*)d_out;                     // [257]

    // Workspace layout (floats)
    float* ws  = (float*)d_ws;
    float* a   = ws;            // [400]   fc1 output
    float* gi1 = ws + 400;      // [1200]
    float* gh1 = ws + 1600;     // [1200]
    float* h1n = ws + 2800;     // [400]
    float* gi2 = ws + 3200;     // [1200]
    float* gh2 = ws + 4400;     // [1200]
    float* h2n = ws + 5600;     // [400]
    float* y1  = ws + 6000;     // [600]
    float* y2  = ws + 6600;     // [600]  (total 7200 floats = 28.8 KB)

    const dim3 blk(256);
    const int dual_blocks = blocks_for_waves(2 * ((1200 + 15) / 16));  // 150 waves

    // fc1: a = fc1_w @ x + fc1_b
    gemv_wmma<0><<<gemv_blocks(400), blk, 0, stream>>>(fc1_w, x, fc1_b, a, 400, 257);

    // GRU1: gi1 = Wi@a, gh1 = Wh@h1 (one launch), then gate combine
    gemv_dual_wmma<<<dual_blocks, blk, 0, stream>>>(gru1_wi, a, gi1,
                                                    gru1_wh, h1, gh1, 1200, 400);
    gru_gate_kernel<<<2, blk, 0, stream>>>(gi1, gh1, gru1_b, h1, h1n);

    // GRU2
    gemv_dual_wmma<<<dual_blocks, blk, 0, stream>>>(gru2_wi, h1n, gi2,
                                                    gru2_wh, h2, gh2, 1200, 400);
    gru_gate_kernel<<<2, blk, 0, stream>>>(gi2, gh2, gru2_b, h2, h2n);

    // fc2 + relu, fc3 + relu, fc4 + sigmoid
    gemv_wmma<1><<<gemv_blocks(600), blk, 0, stream>>>(fc2_w, h2n, fc2_b, y1, 600, 400);
    gemv_wmma<1><<<gemv_blocks(600), blk, 0, stream>>>(fc3_w, y1,  fc3_b, y2, 600, 600);
    gemv_wmma<2><<<gemv_blocks(257), blk, 0, stream>>>(fc4_w, y2,  fc4_b, out, 257, 600);
}